// ChargeIncrementModel_51092930953372
// MI455X (gfx1250) — compile-verified
//
#include <hip/hip_runtime.h>

#define NNODES 50000
#define NEDGES 800000

typedef float v2f __attribute__((ext_vector_type(2)));
typedef float v8f __attribute__((ext_vector_type(8)));

// ---- float <-> monotonic uint encoding for atomicMax on floats ----
__device__ __forceinline__ unsigned encf(float f) {
  unsigned u = __float_as_uint(f);
  return (u & 0x80000000u) ? ~u : (u | 0x80000000u);
}
__device__ __forceinline__ float decf(unsigned u) {
  return (u & 0x80000000u) ? __uint_as_float(u & 0x7FFFFFFFu)
                           : __uint_as_float(~u);
}

// ---- WMMA core: acc(16x64) += A(16 rows x K, lda=K) * W(K x 64 in LDS) ----
// A-frag layout (f32 16x4): lanes0-15 row=l15 K=kk+0..1 ; lanes16-31 row=l15 K=kk+2..3
// B-frag layout (f32 4x16): lanes0-15 col=l15 K=kk+0..1 ; lanes16-31 col=l15 K=kk+2..3
// C/D layout: VGPR r -> row r (lanes0-15) / row r+8 (lanes16-31), col = l15 + nt*16
__device__ __forceinline__ void wmma_rows16_K_64(
    const float* __restrict__ A, int rbase, int K,
    const float* __restrict__ WL, int lane, v8f acc[4])
{
  const int half = lane >> 4;
  const int l15  = lane & 15;
  const float* arow = A + (long long)(rbase + l15) * K + half * 2;
  for (int kk = 0; kk < K; kk += 4) {
    v2f a;
    a.x = arow[kk + 0];
    a.y = arow[kk + 1];
    const float* wr = WL + (kk + half * 2) * 64 + l15;
#pragma unroll
    for (int nt = 0; nt < 4; ++nt) {
      v2f b;
      b.x = wr[nt * 16];
      b.y = wr[64 + nt * 16];
      acc[nt] = __builtin_amdgcn_wmma_f32_16x16x4_f32(
          false, a, false, b, (short)0, acc[nt], false, false);
    }
  }
}

// ---- plain node GEMM: OUT[Nx64] = A[NxK] * W[Kx64] ----
__global__ void __launch_bounds__(256)
k_node_gemm(const float* __restrict__ A, const float* __restrict__ W,
            float* __restrict__ OUT, int ntiles, int K)
{
  __shared__ float WL[64 * 64];
  const int tid = threadIdx.x;
  for (int i = tid; i < K * 64; i += 256) WL[i] = W[i];
  __syncthreads();
  const int lane = tid & 31;
  const int tile = blockIdx.x * 8 + (tid >> 5);
  if (tile >= ntiles) return;
  v8f acc[4];
#pragma unroll
  for (int nt = 0; nt < 4; ++nt)
#pragma unroll
    for (int r = 0; r < 8; ++r) acc[nt][r] = 0.f;
  wmma_rows16_K_64(A, tile * 16, K, WL, lane, acc);
  const int half = lane >> 4, l15 = lane & 15;
  const int rb = tile * 16 + half * 8;
#pragma unroll
  for (int r = 0; r < 8; ++r) {
    float* o = OUT + (long long)(rb + r) * 64 + l15;
#pragma unroll
    for (int nt = 0; nt < 4; ++nt) o[nt * 16] = acc[nt][r];
  }
}

// ---- fused edge kernel: f = leaky(heGEMM + fni[src] + fnj[dst] + bias),
//      he_out = relu(f), e-logits = head-wise dot(f, attn), atomicMax segmax ----
__global__ void __launch_bounds__(256)
k_edge_fused(const float* __restrict__ HE, int K,
             const float* __restrict__ W,
             const float* __restrict__ FNI, const float* __restrict__ FNJ,
             const float* __restrict__ BIAS, const float* __restrict__ ATTN,
             const int* __restrict__ SRC, const int* __restrict__ DST,
             float* __restrict__ HEOUT, float* __restrict__ ELOG,
             unsigned* __restrict__ SEGMAX, int ntiles)
{
  __shared__ float WL[64 * 64];
  const int tid = threadIdx.x;
  for (int i = tid; i < K * 64; i += 256) WL[i] = W[i];
  __syncthreads();
  const int lane = tid & 31;
  const int tile = blockIdx.x * 8 + (tid >> 5);
  if (tile >= ntiles) return;
  v8f acc[4];
#pragma unroll
  for (int nt = 0; nt < 4; ++nt)
#pragma unroll
    for (int r = 0; r < 8; ++r) acc[nt][r] = 0.f;
  wmma_rows16_K_64(HE, tile * 16, K, WL, lane, acc);
  const int half = lane >> 4, l15 = lane & 15;
  float biasc[4], attnc[4];
#pragma unroll
  for (int nt = 0; nt < 4; ++nt) {
    int c = nt * 16 + l15;
    biasc[nt] = BIAS[c];
    attnc[nt] = ATTN[c];
  }
#pragma unroll
  for (int r = 0; r < 8; ++r) {
    const int e = tile * 16 + half * 8 + r;
    const int s = SRC[e], d = DST[e];
    const float* pni = FNI + (long long)s * 64 + l15;
    const float* pnj = FNJ + (long long)d * 64 + l15;
    float* po = HEOUT + (long long)e * 64 + l15;
    float p0 = 0.f, p1 = 0.f;
#pragma unroll
    for (int nt = 0; nt < 4; ++nt) {
      float v = acc[nt][r] + pni[nt * 16] + pnj[nt * 16] + biasc[nt];
      v = v > 0.f ? v : 0.01f * v;          // leaky_relu
      po[nt * 16] = v > 0.f ? v : 0.f;      // relu(f) -> next-layer he
      float pa = v * attnc[nt];
      if (nt < 2) p0 += pa; else p1 += pa;  // head 0 = cols 0..31, head 1 = 32..63
    }
#pragma unroll
    for (int m = 8; m; m >>= 1) {
      p0 += __shfl_xor(p0, m, 16);
      p1 += __shfl_xor(p1, m, 16);
    }
    if (l15 == 0) {
      ELOG[(long long)e * 2 + 0] = p0;
      ELOG[(long long)e * 2 + 1] = p1;
      atomicMax(&SEGMAX[(unsigned)d * 2 + 0], encf(p0));
      atomicMax(&SEGMAX[(unsigned)d * 2 + 1], encf(p1));
    }
  }
}

// ---- edge softmax pass: ex = exp(e - m[dst]); segment-sum ----
__global__ void k_softmax_exp(const int* __restrict__ DST,
                              const unsigned* __restrict__ SEGMAX,
                              float* __restrict__ ELOG,
                              float* __restrict__ SEGSUM, int total)
{
  int i = blockIdx.x * blockDim.x + threadIdx.x;
  if (i >= total) return;
  int e = i >> 1, h = i & 1;
  int d = DST[e];
  float m = decf(SEGMAX[d * 2 + h]);
  float ex = expf(ELOG[i] - m);
  ELOG[i] = ex;
  atomicAdd(&SEGSUM[d * 2 + h], ex);
}

// ---- aggregate: h_out[dst] += h_node[src] * (ex / s[dst]) ----
__global__ void k_aggregate(const int* __restrict__ SRC, const int* __restrict__ DST,
                            const float* __restrict__ ELOG,
                            const float* __restrict__ SEGSUM,
                            const float* __restrict__ HNODE,
                            float* __restrict__ HOUT, int total)
{
  int i = blockIdx.x * blockDim.x + threadIdx.x;
  if (i >= total) return;
  int e = i >> 6, col = i & 63, h = col >> 5;
  int s = SRC[e], d = DST[e];
  float a = ELOG[e * 2 + h] / SEGSUM[d * 2 + h];
  atomicAdd(&HOUT[(long long)d * 64 + col], HNODE[(long long)s * 64 + col] * a);
}

__global__ void k_relu(const float* __restrict__ X, float* __restrict__ Y, int n)
{
  int i = blockIdx.x * blockDim.x + threadIdx.x;
  if (i < n) Y[i] = fmaxf(X[i], 0.f);
}

// ---- fused MLP: inc = selu(he@w0+b0)@w1+b1; out[dst]+=inc, out[src]-=inc ----
__global__ void __launch_bounds__(256)
k_mlp(const float* __restrict__ HE, const float* __restrict__ W0,
      const float* __restrict__ B0, const float* __restrict__ W1,
      const float* __restrict__ B1,
      const int* __restrict__ SRC, const int* __restrict__ DST,
      float* __restrict__ OUT, int ntiles)
{
  __shared__ float WL[64 * 64];
  const int tid = threadIdx.x;
  for (int i = tid; i < 64 * 64; i += 256) WL[i] = W0[i];
  __syncthreads();
  const int lane = tid & 31;
  const int tile = blockIdx.x * 8 + (tid >> 5);
  if (tile >= ntiles) return;
  v8f acc[4];
#pragma unroll
  for (int nt = 0; nt < 4; ++nt)
#pragma unroll
    for (int r = 0; r < 8; ++r) acc[nt][r] = 0.f;
  wmma_rows16_K_64(HE, tile * 16, 64, WL, lane, acc);
  const int half = lane >> 4, l15 = lane & 15;
  float b0c[4], w1c[4];
#pragma unroll
  for (int nt = 0; nt < 4; ++nt) {
    int c = nt * 16 + l15;
    b0c[nt] = B0[c];
    w1c[nt] = W1[c];
  }
  const float sc = 1.0507009873554805f, al = 1.6732632423543772f;
  const float b1 = B1[0];
#pragma unroll
  for (int r = 0; r < 8; ++r) {
    const int e = tile * 16 + half * 8 + r;
    float p = 0.f;
#pragma unroll
    for (int nt = 0; nt < 4; ++nt) {
      float x = acc[nt][r] + b0c[nt];
      x = x > 0.f ? sc * x : sc * al * (expf(x) - 1.f);  // selu
      p += x * w1c[nt];
    }
#pragma unroll
    for (int m = 8; m; m >>= 1) p += __shfl_xor(p, m, 16);
    if (l15 == 0) {
      float inc = p + b1;
      atomicAdd(&OUT[DST[e]], inc);
      atomicAdd(&OUT[SRC[e]], -inc);
    }
  }
}

extern "C" void kernel_launch(void* const* d_in, const int* in_sizes, int n_in,
                              void* d_out, int out_size, void* d_ws, size_t ws_size,
                              hipStream_t stream)
{
  // --- map inputs; disambiguate insertion-order vs alphabetical tree flatten ---
  const float *feats_node, *feats_edge, *charges;
  const int *srcp, *dstp;
  const float *fc_ni[3], *fc_nj[3], *fc_fij[3], *attn[3], *bias[3], *fc_nd[3];
  const float *w0, *b0, *w1, *b1;
  if (in_sizes[0] == NNODES * 64) {  // insertion order
    feats_node = (const float*)d_in[0];
    feats_edge = (const float*)d_in[1];
    charges    = (const float*)d_in[2];
    srcp       = (const int*)d_in[3];
    dstp       = (const int*)d_in[4];
    for (int l = 0; l < 3; ++l) {
      int base = 5 + l * 6;
      fc_ni[l]  = (const float*)d_in[base + 0];
      fc_nj[l]  = (const float*)d_in[base + 1];
      fc_fij[l] = (const float*)d_in[base + 2];
      attn[l]   = (const float*)d_in[base + 3];
      bias[l]   = (const float*)d_in[base + 4];
      fc_nd[l]  = (const float*)d_in[base + 5];
    }
    w0 = (const float*)d_in[23]; b0 = (const float*)d_in[24];
    w1 = (const float*)d_in[25]; b1 = (const float*)d_in[26];
  } else {  // alphabetical (jax tree-flatten) order
    charges    = (const float*)d_in[0];
    dstp       = (const int*)d_in[1];
    feats_edge = (const float*)d_in[2];
    feats_node = (const float*)d_in[3];
    for (int l = 0; l < 3; ++l) {
      int base = 4 + l * 6;
      attn[l]   = (const float*)d_in[base + 0];
      bias[l]   = (const float*)d_in[base + 1];
      fc_fij[l] = (const float*)d_in[base + 2];
      fc_ni[l]  = (const float*)d_in[base + 3];
      fc_nj[l]  = (const float*)d_in[base + 4];
      fc_nd[l]  = (const float*)d_in[base + 5];
    }
    b0 = (const float*)d_in[22]; b1 = (const float*)d_in[23];
    w0 = (const float*)d_in[24]; w1 = (const float*)d_in[25];
    srcp = (const int*)d_in[26];
  }

  // --- workspace carve-up ---
  char* ws = (char*)d_ws;
  size_t off = 0;
  auto alloc = [&](size_t bytes) -> void* {
    void* p = ws + off;
    off += (bytes + 255) & ~(size_t)255;
    return p;
  };
  float* hn       = (float*)alloc((size_t)NNODES * 64 * 4);
  float* fni      = (float*)alloc((size_t)NNODES * 64 * 4);
  float* fnj      = (float*)alloc((size_t)NNODES * 64 * 4);
  float* hnode    = (float*)alloc((size_t)NNODES * 64 * 4);
  float* hout     = (float*)alloc((size_t)NNODES * 64 * 4);
  float* heA      = (float*)alloc((size_t)NEDGES * 64 * 4);
  float* heB      = (float*)alloc((size_t)NEDGES * 64 * 4);
  float* elog     = (float*)alloc((size_t)NEDGES * 2 * 4);
  unsigned* segmax = (unsigned*)alloc((size_t)NNODES * 2 * 4);
  float* segsum    = (float*)alloc((size_t)NNODES * 2 * 4);

  const int ntilesN = NNODES / 16;   // 3125
  const int ntilesE = NEDGES / 16;   // 50000
  const int gN = (ntilesN + 7) / 8;  // 8 waves / block
  const int gE = (ntilesE + 7) / 8;

  const float* hn_in = feats_node;
  const float* he_in = feats_edge;
  float* he_out = heA;
  int Ke = 32;  // layer0 edge feature dim

  for (int l = 0; l < 3; ++l) {
    hipMemsetAsync(segmax, 0, (size_t)NNODES * 2 * 4, stream);  // enc identity (< any real enc)
    hipMemsetAsync(segsum, 0, (size_t)NNODES * 2 * 4, stream);
    hipMemsetAsync(hout,   0, (size_t)NNODES * 64 * 4, stream);
    k_node_gemm<<<gN, 256, 0, stream>>>(hn_in, fc_ni[l], fni,   ntilesN, 64);
    k_node_gemm<<<gN, 256, 0, stream>>>(hn_in, fc_nj[l], fnj,   ntilesN, 64);
    k_node_gemm<<<gN, 256, 0, stream>>>(hn_in, fc_nd[l], hnode, ntilesN, 64);
    k_edge_fused<<<gE, 256, 0, stream>>>(he_in, Ke, fc_fij[l], fni, fnj,
                                         bias[l], attn[l], srcp, dstp,
                                         he_out, elog, segmax, ntilesE);
    k_softmax_exp<<<(NEDGES * 2 + 255) / 256, 256, 0, stream>>>(
        dstp, segmax, elog, segsum, NEDGES * 2);
    k_aggregate<<<(NEDGES * 64 + 255) / 256, 256, 0, stream>>>(
        srcp, dstp, elog, segsum, hnode, hout, NEDGES * 64);
    k_relu<<<(NNODES * 64 + 255) / 256, 256, 0, stream>>>(hout, hn, NNODES * 64);
    hn_in = hn;
    he_in = he_out;
    he_out = (he_out == heA) ? heB : heA;
    Ke = 64;
  }

  hipMemcpyAsync(d_out, charges, (size_t)NNODES * 4, hipMemcpyDeviceToDevice, stream);
  k_mlp<<<gE, 256, 0, stream>>>(he_in, w0, b0, w1, b1, srcp, dstp,
                                (float*)d_out, ntilesE);
}